// GridFunction_8658654069032
// MI455X (gfx1250) — compile-verified
//
#include <hip/hip_runtime.h>

#define G_PTS 128
#define NTHREADS 256

// ---- gfx1250 async global->LDS path (ASYNCcnt), guarded for portability ----
#if defined(__has_builtin)
# if __has_builtin(__builtin_amdgcn_global_load_async_to_lds_b32)
#  define USE_ASYNC_LDS 1
# else
#  define USE_ASYNC_LDS 0
# endif
#else
# define USE_ASYNC_LDS 0
#endif

typedef __attribute__((address_space(1))) int GlobalInt;
typedef __attribute__((address_space(3))) int LdsInt;

__device__ __forceinline__ void async_cp_b32(const float* src, float* lds_dst) {
#if USE_ASYNC_LDS
    __builtin_amdgcn_global_load_async_to_lds_b32(
        (GlobalInt*)src, (LdsInt*)lds_dst, 0, 0);
#else
    *lds_dst = *src;
#endif
}

__device__ __forceinline__ void async_wait0() {
#if USE_ASYNC_LDS
# if __has_builtin(__builtin_amdgcn_s_wait_asynccnt)
    __builtin_amdgcn_s_wait_asynccnt(0);
# else
    asm volatile("s_wait_asynccnt 0" ::: "memory");
# endif
#endif
}

__device__ __forceinline__ void fma4(float4& a, float w, float4 v) {
    a.x = fmaf(w, v.x, a.x);
    a.y = fmaf(w, v.y, a.y);
    a.z = fmaf(w, v.z, a.z);
    a.w = fmaf(w, v.w, a.w);
}

// Replicates: xq = clip(x, g[0], g[127]); li = searchsorted(g, xq, 'right')-1;
// li -= (xq == g[127]); li = clip(li, 0, 126); t = (xq-g[li])/(g[li+1]-g[li]).
// Branchless binary search for the largest i with g[i] <= xq (xq >= g[0] after
// clamping, so lo=0 is a valid start); lo==127 only when xq==g[127], and
// min(lo,126) then lands in the last cell with t == 1, matching the reference.
__device__ __forceinline__ void search_dim(const float* g, float x,
                                           int& li, float& t) {
    const float g0 = g[0];
    const float gN = g[G_PTS - 1];
    float xq = fminf(fmaxf(x, g0), gN);
    int lo = 0;
#pragma unroll
    for (int s = 64; s >= 1; s >>= 1) {
        if (g[lo + s] <= xq) lo += s;
    }
    li = (lo < G_PTS - 2) ? lo : (G_PTS - 2);
    const float gl = g[li];
    const float gr = g[li + 1];
    t = (xq - gl) / (gr - gl);
}

__global__ void __launch_bounds__(NTHREADS)
trilerp_kernel(const float* __restrict__ x, const float* __restrict__ y,
               const float* __restrict__ xs0, const float* __restrict__ xs1,
               const float* __restrict__ xs2, float* __restrict__ out, int n) {
    __shared__ float s_xs[3 * G_PTS];

    // Stage the three coordinate tables into LDS via the gfx1250 async DMA
    // path (global_load_async_to_lds_b32, ASYNCcnt). Straight-line form:
    // 256 threads cover xs0|xs1 with a branchless pointer select, the low
    // 128 threads cover xs2.
    {
        const int tid  = threadIdx.x;
        const int lane = tid & (G_PTS - 1);
        const float* src01 = (tid < G_PTS) ? (xs0 + lane) : (xs1 + lane);
        async_cp_b32(src01, &s_xs[tid]);
        if (tid < G_PTS) {
            async_cp_b32(xs2 + tid, &s_xs[2 * G_PTS + tid]);
        }
        async_wait0();
    }
    __syncthreads();

    const int q = blockIdx.x * NTHREADS + threadIdx.x;
    if (q >= n) return;

    const float xa = x[3 * q + 0];
    const float xb = x[3 * q + 1];
    const float xc = x[3 * q + 2];

    int i0, i1, i2;
    float t0, t1, t2;
    search_dim(s_xs, xa, i0, t0);
    search_dim(s_xs + G_PTS, xb, i1, t1);
    search_dim(s_xs + 2 * G_PTS, xc, i2, t2);

    const float u0 = 1.0f - t0, u1 = 1.0f - t1, u2 = 1.0f - t2;

    // Cell index in units of 16-float blocks: ((i0*G + i1)*G + i2).
    // i2 and i2+1 corners are contiguous (8 float4 = 128 B per (i0,i1) combo).
    const int c00 = (i0 * G_PTS + i1) * G_PTS + i2;
    const int cells[4] = { c00,
                           c00 + G_PTS,
                           c00 + G_PTS * G_PTS,
                           c00 + G_PTS * G_PTS + G_PTS };
    const float wab[4] = { u0 * u1, u0 * t1, t0 * u1, t0 * t1 };

    const float4* __restrict__ yv = reinterpret_cast<const float4*>(y);

    float4 acc[4];
    acc[0] = make_float4(0.f, 0.f, 0.f, 0.f);
    acc[1] = acc[0];
    acc[2] = acc[0];
    acc[3] = acc[0];

#pragma unroll
    for (int cc = 0; cc < 4; ++cc) {
        const float4* p = yv + (size_t)cells[cc] * 4;
        const float wl = wab[cc] * u2;
        const float wh = wab[cc] * t2;
#pragma unroll
        for (int j = 0; j < 4; ++j) {
            float4 vlo = p[j];
            float4 vhi = p[j + 4];
            fma4(acc[j], wl, vlo);
            fma4(acc[j], wh, vhi);
        }
    }

    float4* ov = reinterpret_cast<float4*>(out) + (size_t)q * 4;
#pragma unroll
    for (int j = 0; j < 4; ++j) ov[j] = acc[j];
}

extern "C" void kernel_launch(void* const* d_in, const int* in_sizes, int n_in,
                              void* d_out, int out_size, void* d_ws, size_t ws_size,
                              hipStream_t stream) {
    const float* x   = (const float*)d_in[0];   // (N, 3)
    const float* y   = (const float*)d_in[1];   // (128,128,128,16)
    const float* xs0 = (const float*)d_in[2];   // (128,)
    const float* xs1 = (const float*)d_in[3];   // (128,)
    const float* xs2 = (const float*)d_in[4];   // (128,)
    float* out = (float*)d_out;                 // (N, 16)

    const int n = in_sizes[0] / 3;
    const int blocks = (n + NTHREADS - 1) / NTHREADS;
    hipLaunchKernelGGL(trilerp_kernel, dim3(blocks), dim3(NTHREADS), 0, stream,
                       x, y, xs0, xs1, xs2, out, n);
    (void)d_ws; (void)ws_size; (void)n_in; (void)out_size;
}